// BatchedTeacherPolicy_8349416424025
// MI455X (gfx1250) — compile-verified
//
#include <hip/hip_runtime.h>
#include <hip/hip_bf16.h>
#include <stdint.h>

typedef __attribute__((ext_vector_type(2))) float v2f;
typedef __attribute__((ext_vector_type(8))) float v8f;
typedef __attribute__((ext_vector_type(4))) unsigned int du4;
typedef __attribute__((ext_vector_type(8))) int di8;
typedef __attribute__((ext_vector_type(4))) int di4;

#define NWAVES   8
#define NTHREADS 256
#define CHUNKF_MAX (16 * 36)  // floats per staged chunk (KC=32, RS=36)

// ---------------------------------------------------------------------------
// TDM: issue a 2D tensor tile load (tile_d1 rows x tile_d0 cols of f32, row
// stride = stride0 elements) from global memory into LDS at lds_byte.
// LDS pad: pad_amt(code 3 -> 4 DWORDs) every 2^pad_ivl 8-byte units
// -> LDS row stride = tile_d0 + 4 floats (bank-conflict-free for A reads).
// D# layout per CDNA5 ISA ch.8 (group0 128b, group1 256b; groups 2/3 zero
// for 2D tensors). clang-23 6-arg builtin form.
// ---------------------------------------------------------------------------
__device__ __forceinline__ void tdm_load_2d(const float* gptr, unsigned lds_byte,
                                            unsigned tensor_d0, unsigned tensor_d1,
                                            unsigned tile_d0, unsigned tile_d1,
                                            unsigned stride0, unsigned pad_ivl,
                                            unsigned pad_amt) {
  const unsigned long long ga = (unsigned long long)(uintptr_t)gptr;
  du4 g0;
  g0.x = 1u;                                                // count=1 (valid D#)
  g0.y = lds_byte;                                          // lds_addr (bytes)
  g0.z = (unsigned)ga;                                      // global_addr[31:0]
  g0.w = ((unsigned)(ga >> 32) & 0x01FFFFFFu) | (2u << 30); // addr[56:32]|type=2
  di8 g1;
  g1[0] = (int)((2u << 16)            // data_size = 4B
                | (1u << 20)          // pad_enable
                | (pad_ivl << 22)     // pad_interval code
                | (pad_amt << 25));   // pad_amount code
  g1[1] = (int)((tensor_d0 & 0xFFFFu) << 16);                       // dim0[15:0]
  g1[2] = (int)(((tensor_d0 >> 16) & 0xFFFFu) |
                ((tensor_d1 & 0xFFFFu) << 16));                     // dim0 hi | dim1 lo
  g1[3] = (int)(((tensor_d1 >> 16) & 0xFFFFu) | (tile_d0 << 16));   // dim1 hi | tile0
  g1[4] = (int)(tile_d1 & 0xFFFFu);                                 // tile1 (tile2=0)
  g1[5] = (int)stride0;                                             // dim0_stride lo
  g1[6] = 0;
  g1[7] = 0;
  di4 z4 = {0, 0, 0, 0};
  di8 z8 = {0, 0, 0, 0, 0, 0, 0, 0};
  __builtin_amdgcn_tensor_load_to_lds(g0, g1, z4, z4, z8, 0);
}

// ---------------------------------------------------------------------------
// Accumulate NTILE 16-row output tiles via WMMA f32 16x16x4, with weights
// TDM-staged into LDS in KC-column chunks (double-buffered).
// A layout: lane l (r=l&15, h=l>>4): VGPR0/1 = W[r][k0+2h], W[r][k0+2h+1]
// B layout: x broadcast across all 16 columns -> every D column = y.
// ---------------------------------------------------------------------------
template <int IN_D, int KC, int NTILE>
__device__ __forceinline__ void mlp_tiles_tdm(const float* __restrict__ Wg,
                                              int rb0, int rb1, int out_d,
                                              const float* __restrict__ xin,
                                              float* __restrict__ stg, int lane,
                                              v8f& acc0, v8f& acc1) {
  constexpr int RS  = KC + 4;         // LDS row stride (floats), conflict-free
  constexpr int CF  = 16 * RS;        // floats per staged chunk
  constexpr int NCH = IN_D / KC;
  constexpr unsigned PIVL = (KC == 32) ? 4u : 3u;  // pad every 32 / 16 DWORDs
  constexpr unsigned PAMT = 3u;                    // pad 4 DWORDs
  const int r = lane & 15;
  const int h = lane >> 4;

  auto issue = [&](int buf, int c) {
    tdm_load_2d(Wg + (size_t)rb0 * IN_D + c * KC,
                (unsigned)(uintptr_t)(stg + (buf * 2) * CF),
                (unsigned)(IN_D - c * KC), (unsigned)(out_d - rb0),
                (unsigned)KC, 16u, (unsigned)IN_D, PIVL, PAMT);
    if (NTILE == 2) {
      tdm_load_2d(Wg + (size_t)rb1 * IN_D + c * KC,
                  (unsigned)(uintptr_t)(stg + (buf * 2 + 1) * CF),
                  (unsigned)(IN_D - c * KC), (unsigned)(out_d - rb1),
                  (unsigned)KC, 16u, (unsigned)IN_D, PIVL, PAMT);
    }
  };

  asm volatile("" ::: "memory");
  issue(0, 0);
  for (int c = 0; c < NCH; ++c) {
    if (c + 1 < NCH) {
      issue((c + 1) & 1, c + 1);                 // prefetch next chunk
      __builtin_amdgcn_s_wait_tensorcnt(NTILE);  // current chunk complete
    } else {
      __builtin_amdgcn_s_wait_tensorcnt(0);
    }
    asm volatile("" ::: "memory");               // order LDS reads after wait
    const float* s0 = stg + ((c & 1) * 2) * CF + r * RS + 2 * h;
    const float* s1 = s0 + CF;
    const float* xp = xin + c * KC + 2 * h;
#pragma unroll
    for (int k0 = 0; k0 < KC; k0 += 4) {
      v2f b  = *(const v2f*)xp;
      v2f a0 = *(const v2f*)s0;
      acc0 = __builtin_amdgcn_wmma_f32_16x16x4_f32(false, a0, false, b, (short)0,
                                                   acc0, false, false);
      if (NTILE == 2) {
        v2f a1 = *(const v2f*)s1;
        acc1 = __builtin_amdgcn_wmma_f32_16x16x4_f32(false, a1, false, b, (short)0,
                                                     acc1, false, false);
      }
      s0 += 4; s1 += 4; xp += 4;
    }
  }
}

// C/D layout: lane 0 holds rows rbase+0..7 (VGPR 0..7), lane 16 rows rbase+8..15.
__device__ __forceinline__ void store_tile_elu(v8f acc, const float* __restrict__ bias,
                                               float* __restrict__ xout, int rbase,
                                               int lane) {
  if ((lane & 15) == 0) {
    const int h = lane >> 4;
#pragma unroll
    for (int v = 0; v < 8; ++v) {
      const int row = rbase + v + 8 * h;
      const float z = acc[v] + bias[row];
      xout[row] = z > 0.0f ? z : (__expf(z) - 1.0f);  // ELU(alpha=1)
    }
  }
}

__global__ __launch_bounds__(NTHREADS, 1)
void teacher_mlp_kernel(const float* __restrict__ obs, const float* __restrict__ mean,
                        const float* __restrict__ stdv,
                        const float* __restrict__ W0, const float* __restrict__ b0,
                        const float* __restrict__ W1, const float* __restrict__ b1,
                        const float* __restrict__ W2, const float* __restrict__ b2,
                        const float* __restrict__ W3, const float* __restrict__ b3,
                        float* __restrict__ out) {
  __shared__ __align__(16) float xA[512];
  __shared__ __align__(16) float xB[512];
  __shared__ __align__(16) float stage[NWAVES * 4 * CHUNKF_MAX];  // 72 KB

  const int n    = blockIdx.x;
  const int tid  = threadIdx.x;
  const int lane = tid & 31;
  // Wave id is wave-uniform: scalarize it once so all TDM descriptor and
  // address arithmetic stays in SALU/SGPRs (no per-field v_readfirstlane).
  const int wave = __builtin_amdgcn_readfirstlane(tid >> 5);
  float* st = stage + wave * 4 * CHUNKF_MAX;
  v8f dummy = {};

  // Observation normalization + clamp -> xA[0..47]
  if (tid < 48) {
    const size_t i = (size_t)n * 48 + tid;
    float v = (obs[i] - mean[i]) / stdv[i];
    v = fminf(fmaxf(v, -5.0f), 5.0f);
    xA[tid] = v;
  }
  __syncthreads();

  // Layer 0: 48 -> 512 (32 tiles, 2 pairs/wave), ELU, xA -> xB
  {
    const float* W  = W0 + (size_t)n * 512 * 48;
    const float* bb = b0 + (size_t)n * 512;
#pragma unroll
    for (int p = 0; p < 2; ++p) {
      const int t0 = wave + p * 16, t1 = t0 + 8;
      v8f a0 = {}, a1 = {};
      mlp_tiles_tdm<48, 16, 2>(W, t0 * 16, t1 * 16, 512, xA, st, lane, a0, a1);
      store_tile_elu(a0, bb, xB, t0 * 16, lane);
      store_tile_elu(a1, bb, xB, t1 * 16, lane);
    }
  }
  __syncthreads();

  // Layer 1: 512 -> 256 (16 tiles, 1 pair/wave), ELU, xB -> xA
  {
    const float* W  = W1 + (size_t)n * 256 * 512;
    const float* bb = b1 + (size_t)n * 256;
    v8f a0 = {}, a1 = {};
    mlp_tiles_tdm<512, 32, 2>(W, wave * 16, (wave + 8) * 16, 256, xB, st, lane, a0, a1);
    store_tile_elu(a0, bb, xA, wave * 16, lane);
    store_tile_elu(a1, bb, xA, (wave + 8) * 16, lane);
  }
  __syncthreads();

  // Layer 2: 256 -> 128 (8 tiles, 1/wave), ELU, xA -> xB
  {
    const float* W  = W2 + (size_t)n * 128 * 256;
    const float* bb = b2 + (size_t)n * 128;
    v8f a0 = {};
    mlp_tiles_tdm<256, 32, 1>(W, wave * 16, 0, 128, xA, st, lane, a0, dummy);
    store_tile_elu(a0, bb, xB, wave * 16, lane);
  }
  __syncthreads();

  // Layer 3: 128 -> 12 (single tile, wave 0), tanh, xB -> out.
  // tensor_dim1 = 12 -> TDM zero-fills OOB rows 12..15.
  if (wave == 0) {
    const float* W  = W3 + (size_t)n * 12 * 128;
    const float* bb = b3 + (size_t)n * 12;
    v8f acc = {};
    mlp_tiles_tdm<128, 32, 1>(W, 0, 0, 12, xB, st, lane, acc, dummy);
    if ((lane & 15) == 0) {
      const int h = lane >> 4;
#pragma unroll
      for (int v = 0; v < 8; ++v) {
        const int row = v + 8 * h;
        if (row < 12) {
          out[(size_t)n * 12 + row] = tanhf(acc[v] + bb[row]);
        }
      }
    }
  }
}

extern "C" void kernel_launch(void* const* d_in, const int* in_sizes, int n_in,
                              void* d_out, int out_size, void* d_ws, size_t ws_size,
                              hipStream_t stream) {
  const float* obs  = (const float*)d_in[0];
  const float* mean = (const float*)d_in[1];
  const float* stdv = (const float*)d_in[2];
  const float* W0   = (const float*)d_in[3];
  const float* b0   = (const float*)d_in[4];
  const float* W1   = (const float*)d_in[5];
  const float* b1   = (const float*)d_in[6];
  const float* W2   = (const float*)d_in[7];
  const float* b2   = (const float*)d_in[8];
  const float* W3   = (const float*)d_in[9];
  const float* b3   = (const float*)d_in[10];
  float* out        = (float*)d_out;

  const int N = in_sizes[0] / 48;  // 2048
  teacher_mlp_kernel<<<N, NTHREADS, 0, stream>>>(obs, mean, stdv, W0, b0, W1, b1,
                                                 W2, b2, W3, b3, out);
}